// SGSPAssign2_18537078849982
// MI455X (gfx1250) — compile-verified
//
#include <hip/hip_runtime.h>
#include <hip/hip_bf16.h>

#define NN 50000
#define EE 800000
#define HD 128
#define KK 64

typedef float v2f __attribute__((ext_vector_type(2)));
typedef float v8f __attribute__((ext_vector_type(8)));

// ---------------------------------------------------------------------------
// Edge scatter-add: one wave per edge, each lane moves 4 contiguous floats.
// agg[dst] += feat[src]   (fp32 atomics; all traffic is L2-resident)
// ---------------------------------------------------------------------------
__global__ __launch_bounds__(256) void scatter_add_kernel(
    const float* __restrict__ feat, const int* __restrict__ src,
    const int* __restrict__ dst, float* __restrict__ agg, int E) {
  int t = blockIdx.x * blockDim.x + threadIdx.x;
  int e = t >> 5;
  if (e >= E) return;
  int f = (t & 31) * 4;
  int s = src[e];
  int d = dst[e];
  const float4 v = *(const float4*)(feat + (size_t)s * HD + f);
  float* base = agg + (size_t)d * HD + f;
  atomicAdd(base + 0, v.x);
  atomicAdd(base + 1, v.y);
  atomicAdd(base + 2, v.z);
  atomicAdd(base + 3, v.w);
}

// ---------------------------------------------------------------------------
// T = (1+eps)*x + agg   (elementwise, float4; in-place safe: same index)
// ---------------------------------------------------------------------------
__global__ __launch_bounds__(256) void combine_kernel(
    const float* x, const float* agg, const float* __restrict__ eps_p,
    float* T, int n4) {
  int i = blockIdx.x * blockDim.x + threadIdx.x;
  if (i >= n4) return;
  float c = 1.0f + *eps_p;
  float4 xv = ((const float4*)x)[i];
  float4 av = ((const float4*)agg)[i];
  float4 o;
  o.x = fmaf(c, xv.x, av.x);
  o.y = fmaf(c, xv.y, av.y);
  o.z = fmaf(c, xv.z, av.z);
  o.w = fmaf(c, xv.w, av.w);
  ((float4*)T)[i] = o;
}

// ---------------------------------------------------------------------------
// WMMA fp32 GEMM: C[N,OUT] = A[N,128] @ W[128,OUT] + bias, optional ReLU,
// optional per-column sum / sum-of-squares accumulation (for BatchNorm).
//
// Each wave owns TWO 16-row tiles (M=32): every B fragment fetched from LDS
// feeds two independent V_WMMA_F32_16X16X4_F32 chains (halves LDS traffic,
// hides the WMMA->WMMA RAW hazard). Weights live in a 64KB LDS tile with an
// XOR bank swizzle (col ^ ((row&2)<<4)) so the two half-wave rows (kk vs
// kk+2) hit disjoint LDS bank sets; the b.x/b.y pair stays OUT*4 bytes apart
// so ds_load_2addr still applies. Remainder tile handled by clamping the
// second tile's rows (compute-but-don't-store) to keep EXEC all-ones.
// In-place A==C is safe: a wave reads only its own rows (consumed into
// accumulators before any store) and row tiles partition across waves.
// ---------------------------------------------------------------------------
template <int OUT, bool RELU, bool STATS>
__global__ __launch_bounds__(256) void gemm_gin_kernel(
    const float* A, const float* __restrict__ W,
    const float* __restrict__ bias, float* C,
    float* __restrict__ colsum, float* __restrict__ colsq, int nTiles) {
  __shared__ float sW[HD * OUT];

  const int t = threadIdx.x;
  // cooperative swizzled weight load (float4 granularity; swizzle is
  // constant within an aligned group of 4 columns)
  for (int i = t; i < HD * OUT / 4; i += 256) {
    const int r = (i * 4) / OUT;
    const int c = (i * 4) % OUT;
    const float4 v = ((const float4*)W)[i];
    const int pc = c ^ ((r & 2) << 4);
    *(float4*)(&sW[r * OUT + pc]) = v;
  }
  __syncthreads();

  const int wave = t >> 5;
  const int lane = t & 31;
  const int tile0 = (blockIdx.x * 8 + wave) * 2;
  if (tile0 >= nTiles) return;
  const int tile1 = tile0 + 1;
  const bool has1 = (tile1 < nTiles);

  const int row0 = tile0 * 16;
  const int row1 = (has1 ? tile1 : tile0) * 16; // clamped duplicate if invalid
  const int m = lane & 15;        // A row within tile / B,C column within tile
  const int kh = (lane >> 4) * 2; // K sub-offset for this half-wave (0 or 2)
  constexpr int NT = OUT / 16;

  v8f acc0[NT], acc1[NT];
#pragma unroll
  for (int nt = 0; nt < NT; ++nt) {
    float b = bias ? bias[nt * 16 + m] : 0.0f;
#pragma unroll
    for (int r = 0; r < 8; ++r) {
      acc0[nt][r] = b;
      acc1[nt][r] = b;
    }
  }

  const float* Ap0 = A + (size_t)(row0 + m) * HD + kh;
  const float* Ap1 = A + (size_t)(row1 + m) * HD + kh;

  for (int k = 0; k < HD; k += 4) {
    v2f a0 = *(const v2f*)(Ap0 + k);
    v2f a1 = *(const v2f*)(Ap1 + k);
    const int kk = k + kh;
#pragma unroll
    for (int nt = 0; nt < NT; ++nt) {
      const int pcol = (nt * 16 + m) ^ ((kk & 2) << 4); // bank swizzle
      v2f b;
      b.x = sW[kk * OUT + pcol];       // W[kk][col]
      b.y = sW[(kk + 1) * OUT + pcol]; // W[kk+1][col] (same swizzle)
      acc0[nt] = __builtin_amdgcn_wmma_f32_16x16x4_f32(
          false, a0, false, b, (short)0, acc0[nt], false, false);
      acc1[nt] = __builtin_amdgcn_wmma_f32_16x16x4_f32(
          false, a1, false, b, (short)0, acc1[nt], false, false);
    }
  }

  // C/D layout: VGPR r -> row (r + 8*(lane>>4)), column nt*16 + (lane&15)
  const int roff = (lane >> 4) * 8;
#pragma unroll
  for (int nt = 0; nt < NT; ++nt) {
    const int col = nt * 16 + m;
    float s1 = 0.0f, s2 = 0.0f;
#pragma unroll
    for (int r = 0; r < 8; ++r) {
      float v = acc0[nt][r];
      if (RELU) v = fmaxf(v, 0.0f);
      C[(size_t)(row0 + roff + r) * OUT + col] = v;
      if (STATS) {
        s1 += v;
        s2 += v * v;
      }
    }
    if (has1) {
#pragma unroll
      for (int r = 0; r < 8; ++r) {
        float v = acc1[nt][r];
        if (RELU) v = fmaxf(v, 0.0f);
        C[(size_t)(row1 + roff + r) * OUT + col] = v;
        if (STATS) {
          s1 += v;
          s2 += v * v;
        }
      }
    }
    if (STATS) {
      atomicAdd(&colsum[col], s1);
      atomicAdd(&colsq[col], s2);
    }
  }
}

// ---------------------------------------------------------------------------
// BatchNorm params from column sums: scale = gamma*rsqrt(var+eps),
// shift = beta - mean*scale (training-mode, biased variance)
// ---------------------------------------------------------------------------
__global__ void bn_params_kernel(const float* __restrict__ colsum,
                                 const float* __restrict__ colsq,
                                 const float* __restrict__ gamma,
                                 const float* __restrict__ beta,
                                 float* __restrict__ scale,
                                 float* __restrict__ shift, float invN) {
  int c = threadIdx.x;
  float mean = colsum[c] * invN;
  float var = colsq[c] * invN - mean * mean;
  float s = gamma[c] * rsqrtf(var + 1e-5f);
  scale[c] = s;
  shift[c] = beta[c] - mean * s;
}

// h = relu(h*scale[col] + shift[col]), in place, float4
__global__ __launch_bounds__(256) void bn_apply_relu_kernel(
    float* h, const float* __restrict__ scale, const float* __restrict__ shift,
    int n4) {
  int i = blockIdx.x * blockDim.x + threadIdx.x;
  if (i >= n4) return;
  int col = (i * 4) & (HD - 1);
  float4 v = ((const float4*)h)[i];
  v.x = fmaxf(fmaf(v.x, scale[col + 0], shift[col + 0]), 0.0f);
  v.y = fmaxf(fmaf(v.y, scale[col + 1], shift[col + 1]), 0.0f);
  v.z = fmaxf(fmaf(v.z, scale[col + 2], shift[col + 2]), 0.0f);
  v.w = fmaxf(fmaf(v.w, scale[col + 3], shift[col + 3]), 0.0f);
  ((float4*)h)[i] = v;
}

// ---------------------------------------------------------------------------
// Softmax over K=64 with temperature tau = exp(log_tau); one wave per row.
// ---------------------------------------------------------------------------
__global__ __launch_bounds__(256) void softmax64_kernel(
    const float* __restrict__ logits, const float* __restrict__ log_tau,
    float* __restrict__ S, int nrows) {
  int row = (blockIdx.x * blockDim.x + threadIdx.x) >> 5;
  int lane = threadIdx.x & 31;
  if (row >= nrows) return;
  float inv_tau = expf(-(*log_tau));
  const float* rp = logits + (size_t)row * KK;
  float a = rp[lane];
  float b = rp[lane + 32];
  float mx = fmaxf(a, b);
#pragma unroll
  for (int o = 16; o > 0; o >>= 1) mx = fmaxf(mx, __shfl_xor(mx, o, 32));
  float ea = expf((a - mx) * inv_tau);
  float eb = expf((b - mx) * inv_tau);
  float s = ea + eb;
#pragma unroll
  for (int o = 16; o > 0; o >>= 1) s += __shfl_xor(s, o, 32);
  float inv = 1.0f / s;
  float* op = S + (size_t)row * KK;
  op[lane] = ea * inv;
  op[lane + 32] = eb * inv;
}

extern "C" void kernel_launch(void* const* d_in, const int* in_sizes, int n_in,
                              void* d_out, int out_size, void* d_ws,
                              size_t ws_size, hipStream_t stream) {
  const float* x = (const float*)d_in[0];
  const int* ei = (const int*)d_in[1]; // [2,E]: src then dst
  const float* W1a = (const float*)d_in[2];
  const float* b1a = (const float*)d_in[3];
  const float* W1b = (const float*)d_in[4];
  const float* b1b = (const float*)d_in[5];
  const float* eps1 = (const float*)d_in[6];
  const float* g1 = (const float*)d_in[7];
  const float* be1 = (const float*)d_in[8];
  const float* W2a = (const float*)d_in[9];
  const float* b2a = (const float*)d_in[10];
  const float* W2b = (const float*)d_in[11];
  const float* b2b = (const float*)d_in[12];
  const float* eps2 = (const float*)d_in[13];
  const float* g2 = (const float*)d_in[14];
  const float* be2 = (const float*)d_in[15];
  const float* Wout = (const float*)d_in[16];
  const float* ltau = (const float*)d_in[17];

  const int* src = ei;
  const int* dst = ei + EE;

  // workspace: hbuf [N,128] | aggbuf [N,128] | stats (512 floats)
  float* hbuf = (float*)d_ws;
  float* aggbuf = hbuf + (size_t)NN * HD;
  float* stats = aggbuf + (size_t)NN * HD;
  float* colsum = stats;
  float* colsq = stats + 128;
  float* scale = stats + 256;
  float* shiftv = stats + 384;

  float* Sout = (float*)d_out;            // [N,64] softmax
  float* logits = Sout + (size_t)NN * KK; // [N,64] logits

  const int nTiles = NN / 16; // 3125
  dim3 gGemm((nTiles + 15) / 16); // 8 waves/block, 2 tiles/wave
  const int scatterBlocks = (EE * 32) / 256;
  const int n4 = NN * HD / 4;
  const int ewBlocks = (n4 + 255) / 256;
  const float invN = 1.0f / (float)NN;

  // ---------------- Layer 1 ----------------
  hipMemsetAsync(aggbuf, 0, (size_t)NN * HD * sizeof(float), stream);
  scatter_add_kernel<<<scatterBlocks, 256, 0, stream>>>(x, src, dst, aggbuf, EE);
  combine_kernel<<<ewBlocks, 256, 0, stream>>>(x, aggbuf, eps1, hbuf, n4);
  gemm_gin_kernel<HD, true, false><<<gGemm, 256, 0, stream>>>(
      hbuf, W1a, b1a, hbuf, nullptr, nullptr, nTiles);
  hipMemsetAsync(colsum, 0, 256 * sizeof(float), stream);
  gemm_gin_kernel<HD, false, true><<<gGemm, 256, 0, stream>>>(
      hbuf, W1b, b1b, hbuf, colsum, colsq, nTiles);
  bn_params_kernel<<<1, 128, 0, stream>>>(colsum, colsq, g1, be1, scale, shiftv, invN);
  bn_apply_relu_kernel<<<ewBlocks, 256, 0, stream>>>(hbuf, scale, shiftv, n4);

  // ---------------- Layer 2 ----------------
  hipMemsetAsync(aggbuf, 0, (size_t)NN * HD * sizeof(float), stream);
  scatter_add_kernel<<<scatterBlocks, 256, 0, stream>>>(hbuf, src, dst, aggbuf, EE);
  combine_kernel<<<ewBlocks, 256, 0, stream>>>(hbuf, aggbuf, eps2, hbuf, n4);
  gemm_gin_kernel<HD, true, false><<<gGemm, 256, 0, stream>>>(
      hbuf, W2a, b2a, hbuf, nullptr, nullptr, nTiles);
  hipMemsetAsync(colsum, 0, 256 * sizeof(float), stream);
  gemm_gin_kernel<HD, false, true><<<gGemm, 256, 0, stream>>>(
      hbuf, W2b, b2b, hbuf, colsum, colsq, nTiles);
  bn_params_kernel<<<1, 128, 0, stream>>>(colsum, colsq, g2, be2, scale, shiftv, invN);
  bn_apply_relu_kernel<<<ewBlocks, 256, 0, stream>>>(hbuf, scale, shiftv, n4);

  // ---------------- Output head ----------------
  gemm_gin_kernel<KK, false, false><<<gGemm, 256, 0, stream>>>(
      hbuf, Wout, nullptr, logits, nullptr, nullptr, nTiles);
  softmax64_kernel<<<(NN * 32 + 255) / 256, 256, 0, stream>>>(logits, ltau, Sout, NN);
}